// CMHSA_57810259804423
// MI455X (gfx1250) — compile-verified
//
#include <hip/hip_runtime.h>
#include <hip/hip_bf16.h>

typedef __attribute__((ext_vector_type(2))) float v2f;
typedef __attribute__((ext_vector_type(4))) float v4f;
typedef __attribute__((ext_vector_type(8))) float v8f;
typedef unsigned int u32x4 __attribute__((ext_vector_type(4)));
typedef int i32x8 __attribute__((ext_vector_type(8)));
typedef int i32x4 __attribute__((ext_vector_type(4)));

// ---------------------------------------------------------------------------
// fp32 WMMA GEMM, 64x64 block tile, 4 waves, each wave owns 2x2 16x16 WMMA
// tiles. K staged through double-buffered LDS in chunks of 32.
// TA: A[m][k] = Ap[k*lda + m]   TB: B[k][n] = Bp[n*ldb + k]
// MODE 0: plain  MODE 1: +bias[m]  MODE 2: s1[z]*acc + s0[z]*vsum[z*N+n]
// ---------------------------------------------------------------------------
#define APAD 36  // 32 + 4: conflict-free strided fragment reads, 16B aligned
#define BPAD 68  // 64 + 4

template <bool TA, bool TB, int MODE>
__global__ __launch_bounds__(128) void gemm_wmma_f32(
    const float* __restrict__ A, const float* __restrict__ B,
    float* __restrict__ D, int M, int N, int K, int lda, int ldb, int ldd,
    long long strideA, long long strideB, long long strideD, float alpha,
    const float* __restrict__ bias, const float* __restrict__ s1s0,
    const float* __restrict__ vsum) {
  __shared__ float Asm[2][64][APAD];
  __shared__ float Bsm[2][32][BPAD];

  const int z = blockIdx.z;
  const float* pA = A + (size_t)z * (size_t)strideA;
  const float* pB = B + (size_t)z * (size_t)strideB;
  float* pD = D + (size_t)z * (size_t)strideD;

  const int tid = threadIdx.x;
  const int wave = tid >> 5;
  const int lane = tid & 31;
  const int hf = lane >> 4;
  const int idx = lane & 15;
  const int wm = (wave >> 1) * 32;  // wave offset in 64x64 block tile
  const int wn = (wave & 1) * 32;

  const int m0 = blockIdx.y * 64;
  const int n0 = blockIdx.x * 64;

  v8f acc[2][2] = {};
  v4f ra[4], rb[4];

  auto load_regs = [&](int k0) {
#pragma unroll
    for (int e = 0; e < 4; ++e) {
      const int q = tid + e * 128;  // 0..511
      if (!TA) {                    // k contiguous: float4 along k
        const int row = q >> 3, kq = (q & 7) << 2;
        ra[e] = *(const v4f*)(pA + (size_t)(m0 + row) * lda + (k0 + kq));
      } else {                      // m contiguous: float4 along m
        const int m4 = (q & 15) << 2, kcol = q >> 4;
        ra[e] = *(const v4f*)(pA + (size_t)(k0 + kcol) * lda + (m0 + m4));
      }
      if (!TB) {                    // n contiguous: float4 along n
        const int krow = q >> 4, nq = (q & 15) << 2;
        rb[e] = *(const v4f*)(pB + (size_t)(k0 + krow) * ldb + (n0 + nq));
      } else {                      // k contiguous: float4 along k
        const int n = q >> 3, kq = (q & 7) << 2;
        rb[e] = *(const v4f*)(pB + (size_t)(n0 + n) * ldb + (k0 + kq));
      }
    }
  };
  auto store_regs = [&](int buf) {
#pragma unroll
    for (int e = 0; e < 4; ++e) {
      const int q = tid + e * 128;
      if (!TA) {
        const int row = q >> 3, kq = (q & 7) << 2;
        *(v4f*)&Asm[buf][row][kq] = ra[e];  // b128, aligned (APAD%4==0)
      } else {
        const int m4 = (q & 15) << 2, kcol = q >> 4;
#pragma unroll
        for (int i = 0; i < 4; ++i) Asm[buf][m4 + i][kcol] = ra[e][i];
      }
      if (!TB) {
        const int krow = q >> 4, nq = (q & 15) << 2;
        *(v4f*)&Bsm[buf][krow][nq] = rb[e];  // b128, aligned (BPAD%4==0)
      } else {
        const int n = q >> 3, kq = (q & 7) << 2;
#pragma unroll
        for (int i = 0; i < 4; ++i) Bsm[buf][kq + i][n] = rb[e][i];
      }
    }
  };

  load_regs(0);
  store_regs(0);
  __syncthreads();

  int cur = 0;
  for (int k0 = 0; k0 < K; k0 += 32) {
    const bool has_next = (k0 + 32) < K;
    if (has_next) load_regs(k0 + 32);  // global loads overlap compute below

#pragma unroll
    for (int kk = 0; kk < 8; ++kk) {
      const int kb = kk * 4 + hf * 2;
      v2f a0, a1, b0, b1;
      a0.x = Asm[cur][wm + idx][kb];
      a0.y = Asm[cur][wm + idx][kb + 1];
      a1.x = Asm[cur][wm + 16 + idx][kb];
      a1.y = Asm[cur][wm + 16 + idx][kb + 1];
      b0.x = Bsm[cur][kb][wn + idx];
      b0.y = Bsm[cur][kb + 1][wn + idx];
      b1.x = Bsm[cur][kb][wn + 16 + idx];
      b1.y = Bsm[cur][kb + 1][wn + 16 + idx];
      acc[0][0] = __builtin_amdgcn_wmma_f32_16x16x4_f32(false, a0, false, b0,
                                                        (short)0, acc[0][0],
                                                        false, false);
      acc[0][1] = __builtin_amdgcn_wmma_f32_16x16x4_f32(false, a0, false, b1,
                                                        (short)0, acc[0][1],
                                                        false, false);
      acc[1][0] = __builtin_amdgcn_wmma_f32_16x16x4_f32(false, a1, false, b0,
                                                        (short)0, acc[1][0],
                                                        false, false);
      acc[1][1] = __builtin_amdgcn_wmma_f32_16x16x4_f32(false, a1, false, b1,
                                                        (short)0, acc[1][1],
                                                        false, false);
    }

    if (has_next) {
      __syncthreads();
      store_regs(cur ^ 1);
      __syncthreads();
      cur ^= 1;
    }
  }

#pragma unroll
  for (int it = 0; it < 2; ++it) {
#pragma unroll
    for (int jt = 0; jt < 2; ++jt) {
      const int gn = n0 + wn + jt * 16 + idx;
#pragma unroll
      for (int j = 0; j < 8; ++j) {
        const int gm = m0 + wm + it * 16 + hf * 8 + j;
        float v = acc[it][jt][j] * alpha;
        if (MODE == 1) v += bias[gm];
        if (MODE == 2)
          v = s1s0[z * 2 + 0] * v + s1s0[z * 2 + 1] * vsum[(size_t)z * N + gn];
        pD[(size_t)gm * ldd + gn] = v;
      }
    }
  }
}

// ---------------------------------------------------------------------------
// Per-row sums of V [B*C rows x T] for the instance-norm constant term
// ---------------------------------------------------------------------------
__global__ __launch_bounds__(256) void vsum_kernel(const float* __restrict__ v,
                                                   float* __restrict__ vsum) {
  __shared__ float red[256];
  const int r = blockIdx.x;
  const int tid = threadIdx.x;
  float s = 0.f;
#pragma unroll
  for (int e = 0; e < 4; ++e) s += v[(size_t)r * 1024 + tid + e * 256];
  red[tid] = s;
  __syncthreads();
  for (int st = 128; st > 0; st >>= 1) {
    if (tid < st) red[tid] += red[tid + st];
    __syncthreads();
  }
  if (tid == 0) vsum[r] = red[0];
}

// ---------------------------------------------------------------------------
// Head-mix + row softmax + sum/sumsq accumulation.
// The 4 head rows for (b,q) are a 2D strided tile: tile 1024x4 with a line
// stride of T*T elements -> staged into LDS with one TDM tensor_load_to_lds.
// grid: (q=1024, g=4, b=4), block 256 (8 waves)
// ---------------------------------------------------------------------------
__global__ __launch_bounds__(256) void headmix_softmax(
    const float* __restrict__ attn_raw, const float* __restrict__ w_head,
    float* __restrict__ a, float* __restrict__ sum_acc,
    float* __restrict__ sumsq_acc) {
  __shared__ float rows[4][1024];  // 16 KB TDM-staged tile
  __shared__ float red[256];
  const int q = blockIdx.x, g = blockIdx.y, b = blockIdx.z;
  const int tid = threadIdx.x;

  float wh[4];
#pragma unroll
  for (int h = 0; h < 4; ++h) wh[h] = w_head[g * 4 + h];

#if __has_builtin(__builtin_amdgcn_tensor_load_to_lds)
  if (tid < 32) {  // wave 0 issues the descriptor (wave-level op, EXEC ignored)
    const unsigned lds_off = (unsigned)(unsigned long long)(void*)&rows[0][0];
    const unsigned long long ga =
        (unsigned long long)(attn_raw + (((size_t)(b * 4)) * 1024 + q) * 1024);
    // ---- D# group 0: count=1 | lds_addr | global_addr | type=2 ----
    u32x4 g0;
    g0[0] = 1u;
    g0[1] = lds_off;
    g0[2] = (unsigned)ga;
    g0[3] = (unsigned)((ga >> 32) & 0x01FFFFFFu) | (2u << 30);
    // ---- D# group 1: data_size=4B, tensor 1024x4, tile 1024x4,
    //      dim0_stride = 2^20 elements (row q of next head) ----
    i32x8 g1;
    g1[0] = 0x00020000;            // workgroup_mask=0, data_size=2 (4 bytes)
    g1[1] = (int)(1024u << 16);    // tensor_dim0[15:0] @ bits 63:48
    g1[2] = (int)(4u << 16);       // tensor_dim0 hi=0 | tensor_dim1[15:0]=4
    g1[3] = (int)(1024u << 16);    // tensor_dim1 hi=0 | tile_dim0=1024
    g1[4] = 4;                     // tile_dim1=4, tile_dim2=0 (2D)
    g1[5] = 0x00100000;            // tensor_dim0_stride[31:0] = 1048576
    g1[6] = 0;                     // stride hi | tensor_dim1_stride lo
    g1[7] = 0;
    i32x4 g2 = (i32x4)0;           // dims 2/3 unused (zeros)
    i32x4 g3 = (i32x4)0;
    i32x8 gx = (i32x8)0;           // extra group (clang-23 6-arg form), unused
    __builtin_amdgcn_tensor_load_to_lds(g0, g1, g2, g3, gx, 0);
    __builtin_amdgcn_s_wait_tensorcnt(0);
  }
  __syncthreads();
#else
#pragma unroll
  for (int h = 0; h < 4; ++h)
#pragma unroll
    for (int e = 0; e < 4; ++e)
      rows[h][tid + e * 256] =
          attn_raw[(((size_t)(b * 4 + h)) * 1024 + q) * 1024 + tid + e * 256];
  __syncthreads();
#endif

  float vals[4];
#pragma unroll
  for (int e = 0; e < 4; ++e) {
    const int t = tid + e * 256;
    float s = 0.f;
#pragma unroll
    for (int h = 0; h < 4; ++h) s += wh[h] * rows[h][t];
    vals[e] = s;
  }
  float m = fmaxf(fmaxf(vals[0], vals[1]), fmaxf(vals[2], vals[3]));
  red[tid] = m;
  __syncthreads();
  for (int st = 128; st > 0; st >>= 1) {
    if (tid < st) red[tid] = fmaxf(red[tid], red[tid + st]);
    __syncthreads();
  }
  const float rowmax = red[0];
  __syncthreads();
  float ls = 0.f;
#pragma unroll
  for (int e = 0; e < 4; ++e) {
    vals[e] = __expf(vals[e] - rowmax);
    ls += vals[e];
  }
  red[tid] = ls;
  __syncthreads();
  for (int st = 128; st > 0; st >>= 1) {
    if (tid < st) red[tid] += red[tid + st];
    __syncthreads();
  }
  const float inv = 1.f / red[0];
  __syncthreads();
  const size_t rowbase = (((size_t)(b * 4 + g)) * 1024 + q) * 1024;
  float lsum = 0.f, lsq = 0.f;
#pragma unroll
  for (int e = 0; e < 4; ++e) {
    const float ae = vals[e] * inv;
    a[rowbase + tid + e * 256] = ae;
    lsum += ae;
    lsq += ae * ae;
  }
  red[tid] = lsum;
  __syncthreads();
  for (int st = 128; st > 0; st >>= 1) {
    if (tid < st) red[tid] += red[tid + st];
    __syncthreads();
  }
  if (tid == 0) atomicAdd(&sum_acc[b * 4 + g], red[0]);
  __syncthreads();
  red[tid] = lsq;
  __syncthreads();
  for (int st = 128; st > 0; st >>= 1) {
    if (tid < st) red[tid] += red[tid + st];
    __syncthreads();
  }
  if (tid == 0) atomicAdd(&sumsq_acc[b * 4 + g], red[0]);
}

// ---------------------------------------------------------------------------
// Finalize instance-norm stats into per-(b,h) affine (s1, s0)
// ---------------------------------------------------------------------------
__global__ void stats_finalize(const float* __restrict__ sum_acc,
                               const float* __restrict__ sumsq_acc,
                               const float* __restrict__ gamma,
                               const float* __restrict__ beta,
                               float* __restrict__ s1s0) {
  const int z = threadIdx.x;
  if (z >= 16) return;
  const float invN = 1.f / (1024.f * 1024.f);
  const float mean = sum_acc[z] * invN;
  const float var = sumsq_acc[z] * invN - mean * mean;
  const float inv = rsqrtf(var + 1e-5f);
  const int g = z & 3;
  const float s1 = gamma[g] * inv;
  s1s0[z * 2 + 0] = s1;
  s1s0[z * 2 + 1] = beta[g] - mean * s1;
}

// ---------------------------------------------------------------------------
extern "C" void kernel_launch(void* const* d_in, const int* in_sizes, int n_in,
                              void* d_out, int out_size, void* d_ws,
                              size_t ws_size, hipStream_t stream) {
  const float* x = (const float*)d_in[0];
  const float* wq = (const float*)d_in[1];
  const float* wk = (const float*)d_in[2];
  const float* wv = (const float*)d_in[3];
  const float* w_head = (const float*)d_in[4];
  const float* gamma = (const float*)d_in[5];
  const float* beta = (const float*)d_in[6];
  const float* w_proj = (const float*)d_in[7];
  const float* b_proj = (const float*)d_in[8];
  float* out = (float*)d_out;

  float* ws = (float*)d_ws;
  float* qb = ws;                     // 1,048,576  [B,C,T]
  float* kb = qb + 1048576;           // 1,048,576
  float* vb = kb + 1048576;           // 1,048,576
  float* attn_raw = vb + 1048576;     // 16,777,216 [B,NH,T,T]
  float* abuf = attn_raw + 16777216;  // 16,777,216 softmaxed
  float* oav = abuf + 16777216;       // 1,048,576  [B,NH,T,d]
  float* vsumb = oav + 1048576;       // 1,024
  float* sacc = vsumb + 1024;         // 16
  float* sqacc = sacc + 16;           // 16
  float* s1s0 = sqacc + 16;           // 32

  (void)hipMemsetAsync(sacc, 0, 64 * sizeof(float), stream);

  const dim3 blk(128);
  // 1) Q/K/V projections: [256x256] @ [256x1024] per batch
  gemm_wmma_f32<false, false, 0><<<dim3(16, 4, 4), blk, 0, stream>>>(
      wq, x, qb, 256, 1024, 256, 256, 1024, 1024, 0, 262144, 262144, 1.0f,
      nullptr, nullptr, nullptr);
  gemm_wmma_f32<false, false, 0><<<dim3(16, 4, 4), blk, 0, stream>>>(
      wk, x, kb, 256, 1024, 256, 256, 1024, 1024, 0, 262144, 262144, 1.0f,
      nullptr, nullptr, nullptr);
  gemm_wmma_f32<false, false, 0><<<dim3(16, 4, 4), blk, 0, stream>>>(
      wv, x, vb, 256, 1024, 256, 256, 1024, 1024, 0, 262144, 262144, 1.0f,
      nullptr, nullptr, nullptr);

  // 2) attn_raw = scale * Q^T K per (b,h): M=N=1024, K=64
  gemm_wmma_f32<true, false, 0><<<dim3(16, 16, 16), blk, 0, stream>>>(
      qb, kb, attn_raw, 1024, 1024, 64, 1024, 1024, 1024, 65536, 65536,
      1048576, 0.125f, nullptr, nullptr, nullptr);

  // 3) V row sums
  vsum_kernel<<<1024, 256, 0, stream>>>(vb, vsumb);

  // 4) head-mix + softmax + stats (TDM-staged head rows)
  headmix_softmax<<<dim3(1024, 4, 4), 256, 0, stream>>>(attn_raw, w_head, abuf,
                                                        sacc, sqacc);

  // 5) finalize instance-norm affine
  stats_finalize<<<1, 16, 0, stream>>>(sacc, sqacc, gamma, beta, s1s0);

  // 6) out_av = s1*(A @ V^T) + s0*vsum : M=1024, N=64, K=1024
  gemm_wmma_f32<false, true, 2><<<dim3(1, 16, 16), blk, 0, stream>>>(
      abuf, vb, oav, 1024, 64, 1024, 1024, 1024, 64, 1048576, 65536, 65536,
      1.0f, nullptr, s1s0, vsumb);

  // 7) final projection (+bias), writes [B,C,T] = output layout directly
  gemm_wmma_f32<false, true, 1><<<dim3(16, 4, 4), blk, 0, stream>>>(
      w_proj, oav, out, 256, 1024, 256, 256, 256, 1024, 0, 262144, 262144,
      1.0f, b_proj, nullptr, nullptr);
}